// EulerToAffine_74208444940704
// MI455X (gfx1250) — compile-verified
//
#include <hip/hip_runtime.h>

// EulerToAffine: (N,6) f32 -> (N,12) f32, N = 4,194,304.
// Memory-bound streaming kernel: 302 MB total traffic @ 23.3 TB/s ~ 13 us.
// Strategy: 2 rows per thread => 48B in = 3x b128 aligned loads,
//           96B out = 6x b128 aligned stores, all non-temporal.

typedef float v4f __attribute__((ext_vector_type(4)));

__device__ __forceinline__ void euler_row(float ax, float ay, float az,
                                          float t0, float t1, float t2,
                                          v4f& o0, v4f& o1, v4f& o2)
{
    float sx, cx, sy, cy, sz, cz;
    __sincosf(ax, &sx, &cx);   // -> v_sin_f32 / v_cos_f32 (TRANS, co-executes with VALU)
    __sincosf(ay, &sy, &cy);
    __sincosf(az, &sz, &cz);

    const float sxsy = sx * sy;
    const float cxsy = cx * sy;

    // R = Rx * Ry * Rz, closed form; output row layout [Ri0 Ri1 Ri2 ti]
    o0 = (v4f){ cy * cz,
                -cy * sz,
                sy,
                t0 };
    o1 = (v4f){ fmaf(sxsy, cz,  cx * sz),
                fmaf(-sxsy, sz, cx * cz),
                -sx * cy,
                t1 };
    o2 = (v4f){ fmaf(-cxsy, cz, sx * sz),
                fmaf(cxsy, sz,  sx * cz),
                cx * cy,
                t2 };
}

__global__ __launch_bounds__(256)
void euler_to_affine_kernel(const float* __restrict__ in,
                            float* __restrict__ out,
                            int npairs, int nrows)
{
    const int i = blockIdx.x * 256 + threadIdx.x;

    const v4f* __restrict__ vin  = (const v4f*)in;
    v4f* __restrict__       vout = (v4f*)out;

    if (i < npairs) {
        // Prefetch ~192 KB ahead (4096 pairs); speculative => OOB-safe,
        // emits global_prefetch_b8 and warms the streaming pipe.
        __builtin_prefetch((const void*)(vin + 3 * (i + 4096)), 0, 0);

        // 48 B = two rows, 16B-aligned: three non-temporal b128 loads.
        const v4f a0 = __builtin_nontemporal_load(vin + 3 * i + 0);
        const v4f a1 = __builtin_nontemporal_load(vin + 3 * i + 1);
        const v4f a2 = __builtin_nontemporal_load(vin + 3 * i + 2);

        // row 2i   : angles (a0.x, a0.y, a0.z), t (a0.w, a1.x, a1.y)
        // row 2i+1 : angles (a1.z, a1.w, a2.x), t (a2.y, a2.z, a2.w)
        v4f r00, r01, r02, r10, r11, r12;
        euler_row(a0.x, a0.y, a0.z, a0.w, a1.x, a1.y, r00, r01, r02);
        euler_row(a1.z, a1.w, a2.x, a2.y, a2.z, a2.w, r10, r11, r12);

        // 96 B = two output rows, 16B-aligned: six non-temporal b128 stores.
        __builtin_nontemporal_store(r00, vout + 6 * i + 0);
        __builtin_nontemporal_store(r01, vout + 6 * i + 1);
        __builtin_nontemporal_store(r02, vout + 6 * i + 2);
        __builtin_nontemporal_store(r10, vout + 6 * i + 3);
        __builtin_nontemporal_store(r11, vout + 6 * i + 4);
        __builtin_nontemporal_store(r12, vout + 6 * i + 5);
    } else if (i == npairs && (nrows & 1)) {
        // Scalar tail for odd N (not hit for the fixed 4M-row problem,
        // kept for determinism/safety).
        const int r = nrows - 1;
        const float ax = in[6 * r + 0], ay = in[6 * r + 1], az = in[6 * r + 2];
        const float t0 = in[6 * r + 3], t1 = in[6 * r + 4], t2 = in[6 * r + 5];
        v4f o0, o1, o2;
        euler_row(ax, ay, az, t0, t1, t2, o0, o1, o2);
        float* o = out + 12 * r;
        o[0] = o0.x; o[1]  = o0.y; o[2]  = o0.z; o[3]  = o0.w;
        o[4] = o1.x; o[5]  = o1.y; o[6]  = o1.z; o[7]  = o1.w;
        o[8] = o2.x; o[9]  = o2.y; o[10] = o2.z; o[11] = o2.w;
    }
}

extern "C" void kernel_launch(void* const* d_in, const int* in_sizes, int n_in,
                              void* d_out, int out_size, void* d_ws, size_t ws_size,
                              hipStream_t stream)
{
    (void)n_in; (void)out_size; (void)d_ws; (void)ws_size;

    const float* in  = (const float*)d_in[0];
    float*       out = (float*)d_out;

    const int nrows  = in_sizes[0] / 6;   // 4,194,304
    const int npairs = nrows / 2;         // 2,097,152
    const int total  = npairs + (nrows & 1);

    const int block = 256;                 // 8 wave32 per block
    const int grid  = (total + block - 1) / block;

    euler_to_affine_kernel<<<grid, block, 0, stream>>>(in, out, npairs, nrows);
}